// SummationMPNN_19868518711817
// MI455X (gfx1250) — compile-verified
//
#include <hip/hip_runtime.h>
#include <hip/hip_bf16.h>
#include <math.h>

// Problem constants (from reference)
static constexpr int kB   = 256;
static constexpr int kN   = 32;
static constexpr int kF   = 40;
static constexpr int kH   = 100;
static constexpr int kM   = 100;
static constexpr int kEF  = 4;
static constexpr int kOUT = 128;
static constexpr int kR   = kB * kN;      // 8192 node rows
static constexpr int PASSES = 3;

// Padded dims (all GEMM operands live in workspace; padding is zero-filled)
static constexpr int Hp  = 128;           // H=100  -> 128
static constexpr int G3p = 320;           // 3H=300 -> 320
static constexpr int SKp = kEF * Hp;      // 512
static constexpr int CKp = 160;           // H+F=140 -> 160

typedef __attribute__((ext_vector_type(2))) float v2f;
typedef __attribute__((ext_vector_type(8))) float v8f;

// ---------------------------------------------------------------------------
// CDNA5 async global->LDS copy (ASYNCcnt-tracked), 16B per lane.
// VDST operand = per-lane LDS byte offset; flat shared pointer low 32 bits
// are the workgroup-relative LDS offset (ISA 10.2 aperture rule).
// ---------------------------------------------------------------------------
__device__ __forceinline__ void async_copy_b128(const void* lds_dst, const float* gsrc)
{
    uint32_t loff = (uint32_t)(uintptr_t)lds_dst;
    uint64_t gaddr = (uint64_t)(uintptr_t)gsrc;
    asm volatile("global_load_async_to_lds_b128 %0, %1, off"
                 :: "v"(loff), "v"(gaddr) : "memory");
}
__device__ __forceinline__ void wait_async0()
{
    asm volatile("s_wait_asynccnt 0x0" ::: "memory");
}
__device__ __forceinline__ void wait_ds0()
{
    asm volatile("s_wait_dscnt 0x0" ::: "memory");
}

// ---------------------------------------------------------------------------
// Guard-free fp32 WMMA GEMM: C(R x Nc) = A(R x K) @ Bt(Nc x K)^T [+ bias]
// A row-major lda=K; Bt is B TRANSPOSED, row-major ldbt=K; C row-major ldc=Nc.
// Requirements: R % 64 == 0, K % 32 == 0, Nc % 64 == 0; 16B-aligned pointers.
// 256 threads = 8 waves; WG tile 64x64; each wave 16x32 (2 accumulators).
// Double-buffered LDS, async global->LDS staging, 1 barrier per K-chunk.
// ---------------------------------------------------------------------------
#define GEMM_BM 64
#define GEMM_BN 64
#define GEMM_KC 32
#define LDS_LD  36   // floats; 144B rows: 16B aligned, conflict-free fragments

__global__ __launch_bounds__(256) void k_gemm_f32_wmma(
    const float* __restrict__ A, const float* __restrict__ Bt,
    float* __restrict__ C, const float* __restrict__ bias,
    int K, int Nc)
{
    __shared__ float As[2][GEMM_BM * LDS_LD];
    __shared__ float Bs[2][GEMM_BM * LDS_LD];   // 64 output-cols x 32 k

    const int tid  = threadIdx.x;
    const int lane = tid & 31;
    const int wave = tid >> 5;        // 0..7
    const int wtr  = wave & 3;        // row tile 0..3 (16 rows each)
    const int wtc  = wave >> 2;       // col group 0..1 (32 cols each)
    const int half = lane >> 4;       // 0: lanes 0-15, 1: lanes 16-31
    const int l16  = lane & 15;

    const int rowBase = blockIdx.y * GEMM_BM;
    const int colBase = blockIdx.x * GEMM_BN;

    // Per-thread staging coordinates (both tiles are 64x32, 512 float4)
    const int sr0 = tid >> 3;          // rows 0..31   (8 float4 per row)
    const int sc0 = (tid & 7) * 4;     // k-offset within chunk
    const int sr1 = sr0 + 32;          // rows 32..63

    auto stage = [&](int k0, int buf) {
        async_copy_b128(&As[buf][sr0 * LDS_LD + sc0],
                        A + (size_t)(rowBase + sr0) * K + k0 + sc0);
        async_copy_b128(&As[buf][sr1 * LDS_LD + sc0],
                        A + (size_t)(rowBase + sr1) * K + k0 + sc0);
        async_copy_b128(&Bs[buf][sr0 * LDS_LD + sc0],
                        Bt + (size_t)(colBase + sr0) * K + k0 + sc0);
        async_copy_b128(&Bs[buf][sr1 * LDS_LD + sc0],
                        Bt + (size_t)(colBase + sr1) * K + k0 + sc0);
    };

    v8f acc0 = {}, acc1 = {};

    const int nChunks = K / GEMM_KC;
    stage(0, 0);

    const int arow  = ((wtr << 4) + l16) * LDS_LD;
    const int brow0 = ((wtc << 5) + l16) * LDS_LD;
    const int brow1 = brow0 + 16 * LDS_LD;

    for (int i = 0; i < nChunks; ++i) {
        wait_async0();      // chunk i resident in LDS (this wave's issues)
        wait_ds0();         // all our LDS reads of the other buffer retired
        __syncthreads();    // whole WG: data ready, other buffer reusable
        if (i + 1 < nChunks) stage((i + 1) * GEMM_KC, (i + 1) & 1);

        const float* as = As[i & 1];
        const float* bs = Bs[i & 1];
        // fp32 WMMA fragments: lanes 0-15 hold K={kk,kk+1}, 16-31 K={kk+2,kk+3}
        #pragma unroll
        for (int kk = 0; kk < GEMM_KC; kk += 4) {
            v2f a, b0, b1;
            a.x  = as[arow  + kk + 2 * half];
            a.y  = as[arow  + kk + 2 * half + 1];
            b0.x = bs[brow0 + kk + 2 * half];
            b0.y = bs[brow0 + kk + 2 * half + 1];
            b1.x = bs[brow1 + kk + 2 * half];
            b1.y = bs[brow1 + kk + 2 * half + 1];
            acc0 = __builtin_amdgcn_wmma_f32_16x16x4_f32(
                false, a, false, b0, (short)0, acc0, false, false);
            acc1 = __builtin_amdgcn_wmma_f32_16x16x4_f32(
                false, a, false, b1, (short)0, acc1, false, false);
        }
    }

    // C layout: VGPR v -> M = v (lanes 0-15) / v+8 (lanes 16-31), N = lane%16
    const int col0 = colBase + (wtc << 5) + l16;
    const float bv0 = bias ? bias[col0]      : 0.0f;
    const float bv1 = bias ? bias[col0 + 16] : 0.0f;
    #pragma unroll
    for (int v = 0; v < 8; ++v) {
        int row = rowBase + (wtr << 4) + v + (half << 3);
        C[(size_t)row * Nc + col0]      = acc0[v] + bv0;
        C[(size_t)row * Nc + col0 + 16] = acc1[v] + bv1;
    }
}

// ---------------------------------------------------------------------------
// Edge types: tR[b*N+n] = one-hot index of edges[b, n, (n+1)%N, :]
// ---------------------------------------------------------------------------
__global__ __launch_bounds__(256) void k_edge_types(
    const float* __restrict__ edges, int* __restrict__ tR)
{
    int r = blockIdx.x * blockDim.x + threadIdx.x;
    if (r >= kR) return;
    int b = r / kN, n = r % kN;
    int j = (n + 1) % kN;
    const float* e = edges + (((size_t)b * kN + n) * kN + j) * kEF;
    int t = 0;
    #pragma unroll
    for (int f = 1; f < kEF; ++f) if (e[f] > 0.5f) t = f;
    tR[r] = t;
}

// W2tT (Hp x SKp): W2tT[m][t*Hp+h] = W_msg[m,h,t]  (B-transposed, zero padded)
__global__ __launch_bounds__(256) void k_wmsg_pack(
    const float* __restrict__ Wmsg, float* __restrict__ W2tT)
{
    int idx = blockIdx.x * blockDim.x + threadIdx.x;
    if (idx >= Hp * SKp) return;
    int m = idx / SKp;
    int k = idx % SKp;
    int t = k / Hp;
    int h = k % Hp;
    W2tT[idx] = (m < kM && h < kH) ? Wmsg[((size_t)m * kH + h) * kEF + t] : 0.0f;
}

// Padded copy: out(Rp x Cp) from in(Rin x Cin), zero filled
__global__ __launch_bounds__(256) void k_pad_copy(
    float* __restrict__ out, const float* __restrict__ in,
    int Rin, int Cin, int Rp, int Cp)
{
    int idx = blockIdx.x * blockDim.x + threadIdx.x;
    if (idx >= Rp * Cp) return;
    int r = idx / Cp, c = idx % Cp;
    out[idx] = (r < Rin && c < Cin) ? in[(size_t)r * Cin + c] : 0.0f;
}

// Padded transpose: out(Ro x Co) with out[r][c] = in[c][r]; in is (InR x InC)
__global__ __launch_bounds__(256) void k_pad_transpose(
    float* __restrict__ out, const float* __restrict__ in,
    int Ro, int Co, int InR, int InC)
{
    int idx = blockIdx.x * blockDim.x + threadIdx.x;
    if (idx >= Ro * Co) return;
    int r = idx / Co, c = idx % Co;
    out[idx] = (c < InR && r < InC) ? in[(size_t)c * InC + r] : 0.0f;
}

// hidden (kR x Hp): cols [0,F) = nodes, rest zero
__global__ __launch_bounds__(256) void k_init_hidden(
    const float* __restrict__ nodes, float* __restrict__ hidden)
{
    int idx = blockIdx.x * blockDim.x + threadIdx.x;
    if (idx >= kR * Hp) return;
    int r = idx / Hp, h = idx % Hp;
    hidden[idx] = (h < kF) ? nodes[(size_t)r * kF + h] : 0.0f;
}

// S (kR x SKp): S[r][t*Hp+h] = (t==tL)*hidden[left][h] + (t==tR)*hidden[right][h]
__global__ __launch_bounds__(256) void k_build_S(
    const float* __restrict__ hidden, const int* __restrict__ tR,
    float* __restrict__ S)
{
    int idx = blockIdx.x * blockDim.x + threadIdx.x;
    if (idx >= kR * Hp) return;
    int r = idx / Hp, h = idx % Hp;
    int b = r / kN, n = r % kN;
    int left  = b * kN + (n + kN - 1) % kN;
    int right = b * kN + (n + 1) % kN;
    int tl = tR[left];   // edge (n-1, n) carries type tR of node n-1
    int tr = tR[r];      // edge (n, n+1)
    float hl = hidden[(size_t)left  * Hp + h];
    float hr = hidden[(size_t)right * Hp + h];
    float* Sr = S + (size_t)r * SKp;
    #pragma unroll
    for (int t = 0; t < kEF; ++t)
        Sr[t * Hp + h] = (t == tl ? hl : 0.0f) + (t == tr ? hr : 0.0f);
}

__device__ __forceinline__ float sigmoidf_(float x) {
    return 1.0f / (1.0f + expf(-x));
}

// Fused GRU cell update; gi/gh row stride G3p, hidden row stride Hp
__global__ __launch_bounds__(256) void k_gru(
    const float* __restrict__ gi, const float* __restrict__ gh,
    const float* __restrict__ bih, const float* __restrict__ bhh,
    float* __restrict__ hidden)
{
    int idx = blockIdx.x * blockDim.x + threadIdx.x;
    if (idx >= kR * kH) return;
    int r = idx / kH, c = idx % kH;
    const float* gir = gi + (size_t)r * G3p;
    const float* ghr = gh + (size_t)r * G3p;
    float ir  = gir[c]          + bih[c];
    float iz  = gir[kH + c]     + bih[kH + c];
    float in_ = gir[2 * kH + c] + bih[2 * kH + c];
    float hr_ = ghr[c]          + bhh[c];
    float hz  = ghr[kH + c]     + bhh[kH + c];
    float hn  = ghr[2 * kH + c] + bhh[2 * kH + c];
    float rg = sigmoidf_(ir + hr_);
    float zg = sigmoidf_(iz + hz);
    float ng = tanhf(in_ + rg * hn);
    size_t hi = (size_t)r * Hp + c;
    float h = hidden[hi];
    hidden[hi] = (1.0f - zg) * ng + zg * h;
}

// cat (kR x CKp): [hidden(100), nodes(40), zeros(20)]
__global__ __launch_bounds__(256) void k_build_cat(
    const float* __restrict__ hidden, const float* __restrict__ nodes,
    float* __restrict__ cat)
{
    int idx = blockIdx.x * blockDim.x + threadIdx.x;
    if (idx >= kR * CKp) return;
    int r = idx / CKp, c = idx % CKp;
    float v = 0.0f;
    if (c < kH)           v = hidden[(size_t)r * Hp + c];
    else if (c < kH + kF) v = nodes[(size_t)r * kF + (c - kH)];
    cat[idx] = v;
}

// gemb (kB x Hp): cols [0,H): sum_n sigmoid(glin+bg)*(elin+be); cols [H,Hp)=0
__global__ __launch_bounds__(256) void k_gate_reduce(
    const float* __restrict__ glin, const float* __restrict__ elin,
    const float* __restrict__ bg, const float* __restrict__ be,
    float* __restrict__ gemb)
{
    int idx = blockIdx.x * blockDim.x + threadIdx.x;
    if (idx >= kB * Hp) return;
    int b = idx / Hp, c = idx % Hp;
    float acc = 0.0f;
    if (c < kH) {
        float bgc = bg[c], bec = be[c];
        for (int n = 0; n < kN; ++n) {
            size_t r = (size_t)(b * kN + n) * Hp + c;
            acc += sigmoidf_(glin[r] + bgc) * (elin[r] + bec);
        }
    }
    gemb[idx] = acc;
}

// ---------------------------------------------------------------------------
static inline dim3 grid1d(int count) { return dim3((count + 255) / 256); }

static inline void launch_gemm(const float* A, const float* Bt, float* C,
                               const float* bias, int R, int K, int Nc,
                               hipStream_t stream)
{
    dim3 grid(Nc / GEMM_BN, R / GEMM_BM);
    k_gemm_f32_wmma<<<grid, 256, 0, stream>>>(A, Bt, C, bias, K, Nc);
}

extern "C" void kernel_launch(void* const* d_in, const int* in_sizes, int n_in,
                              void* d_out, int out_size, void* d_ws, size_t ws_size,
                              hipStream_t stream)
{
    (void)in_sizes; (void)n_in; (void)out_size; (void)ws_size;

    const float* nodes = (const float*)d_in[0];   // (B,N,F)
    const float* edges = (const float*)d_in[1];   // (B,N,N,EF)
    const float* W_msg = (const float*)d_in[2];   // (M,H,EF)
    const float* W_ih  = (const float*)d_in[3];   // (3H,M)
    const float* W_hh  = (const float*)d_in[4];   // (3H,H)
    const float* b_ih  = (const float*)d_in[5];   // (3H,)
    const float* b_hh  = (const float*)d_in[6];   // (3H,)
    const float* Wg    = (const float*)d_in[7];   // (H+F,H)
    const float* bg    = (const float*)d_in[8];   // (H,)
    const float* We    = (const float*)d_in[9];   // (H,H)
    const float* be    = (const float*)d_in[10];  // (H,)
    const float* Wo    = (const float*)d_in[11];  // (H,OUT)
    const float* bo    = (const float*)d_in[12];  // (OUT,)
    float* out = (float*)d_out;                   // (B,OUT) = 256x128

    // Workspace arena (capture-safe; everything fully written before read)
    char* base = (char*)d_ws;
    size_t off = 0;
    auto take = [&](size_t bytes) -> void* {
        void* p = base + off;
        off += (bytes + 255) & ~(size_t)255;
        return p;
    };
    int*   tR     = (int*)  take((size_t)kR * sizeof(int));
    // All B operands stored TRANSPOSED: (Nc x K), row stride = K
    float* W2tT   = (float*)take((size_t)Hp   * SKp * 4);  // 128 x 512
    float* Wih_p  = (float*)take((size_t)G3p  * Hp  * 4);  // 320 x 128 (pad copy)
    float* Whh_p  = (float*)take((size_t)G3p  * Hp  * 4);  // 320 x 128 (pad copy)
    float* Wg_p   = (float*)take((size_t)Hp   * CKp * 4);  // 128 x 160 (transpose)
    float* We_p   = (float*)take((size_t)Hp   * Hp  * 4);  // 128 x 128 (transpose)
    float* Wo_p   = (float*)take((size_t)kOUT * Hp  * 4);  // 128 x 128 (transpose)
    float* hidden = (float*)take((size_t)kR * Hp  * 4);    // 8192 x 128
    float* S      = (float*)take((size_t)kR * SKp * 4);    // 8192 x 512
    float* msg    = (float*)take((size_t)kR * Hp  * 4);    // 8192 x 128
    float* gi     = (float*)take((size_t)kR * G3p * 4);    // 8192 x 320
    float* gh     = (float*)take((size_t)kR * G3p * 4);    // 8192 x 320
    float* gemb   = (float*)take((size_t)kB * Hp  * 4);    // 256 x 128
    float* cat  = S;     // reuse after passes (8192x160 <= 8192x512)
    float* glin = msg;   // reuse (8192x128)
    float* elin = gi;    // reuse (8192x128 <= 8192x320)

    // Preprocessing: edge types + padded (transposed) weights
    k_edge_types<<<grid1d(kR), 256, 0, stream>>>(edges, tR);
    k_wmsg_pack<<<grid1d(Hp * SKp), 256, 0, stream>>>(W_msg, W2tT);
    k_pad_copy<<<grid1d(G3p * Hp), 256, 0, stream>>>(Wih_p, W_ih, 3 * kH, kM, G3p, Hp);
    k_pad_copy<<<grid1d(G3p * Hp), 256, 0, stream>>>(Whh_p, W_hh, 3 * kH, kH, G3p, Hp);
    k_pad_transpose<<<grid1d(Hp * CKp), 256, 0, stream>>>(Wg_p, Wg, Hp, CKp, kH + kF, kH);
    k_pad_transpose<<<grid1d(Hp * Hp), 256, 0, stream>>>(We_p, We, Hp, Hp, kH, kH);
    k_pad_transpose<<<grid1d(kOUT * Hp), 256, 0, stream>>>(Wo_p, Wo, kOUT, Hp, kH, kOUT);
    k_init_hidden<<<grid1d(kR * Hp), 256, 0, stream>>>(nodes, hidden);

    // Message passes
    for (int p = 0; p < PASSES; ++p) {
        k_build_S<<<grid1d(kR * Hp), 256, 0, stream>>>(hidden, tR, S);
        launch_gemm(S,      W2tT,  msg, nullptr, kR, SKp, Hp,  stream);
        launch_gemm(msg,    Wih_p, gi,  nullptr, kR, Hp,  G3p, stream);
        launch_gemm(hidden, Whh_p, gh,  nullptr, kR, Hp,  G3p, stream);
        k_gru<<<grid1d(kR * kH), 256, 0, stream>>>(gi, gh, b_ih, b_hh, hidden);
    }

    // Readout
    k_build_cat<<<grid1d(kR * CKp), 256, 0, stream>>>(hidden, nodes, cat);
    launch_gemm(cat,    Wg_p, glin, nullptr, kR, CKp, Hp, stream);
    launch_gemm(hidden, We_p, elin, nullptr, kR, Hp,  Hp, stream);
    k_gate_reduce<<<grid1d(kB * Hp), 256, 0, stream>>>(glin, elin, bg, be, gemb);
    launch_gemm(gemb, Wo_p, out, bo, kB, Hp, kOUT, stream);
}